// RWKV_RNN_5669356833490
// MI455X (gfx1250) — compile-verified
//
#include <hip/hip_runtime.h>
#include <math.h>

#define L 12
#define D 1024
#define H 4096
#define V 50277
#define LN_EPS 1e-5f

typedef __attribute__((ext_vector_type(2))) float v2f;
typedef __attribute__((ext_vector_type(8))) float v8f;

// Epilogue modes: 0 = y, 1 = relu(y)^2, 2 = out += y,
//                 3 = out = (out + sigmoid(aux)*y) * scale  (FFN tail)
__device__ __forceinline__ void apply_epilogue(float* out, const float* aux,
                                               int r, float y, int mode, float scale)
{
    if (mode == 0) {
        out[r] = y;
    } else if (mode == 1) {
        float t = fmaxf(y, 0.0f);
        out[r] = t * t;
    } else if (mode == 2) {
        out[r] += y;
    } else {
        float s = 1.0f / (1.0f + expf(-aux[r]));
        out[r] = (out[r] + s * y) * scale;
    }
}

// ---------------------------------------------------------------------------
// WMMA f32 matvec: each wave computes 16 output rows of  y = W[nrows x K] @ x
// over one K-chunk (blockIdx.y selects chunk -> partial sums; a combine kernel
// reduces chunks). blockIdx.z batches up to 3 independent matvecs (k/v/r).
// A tile (16x4 fp32): one global_load_b64 per lane matches the ISA 32-bit A
// 16x4 layout (lanes 0-15 -> K={0,1}, lanes 16-31 -> K={2,3}). B broadcasts x
// into all 16 columns (ds_load_b64 with the same col addressing), so D
// replicates y across N; lane0 holds y[0..7], lane16 holds y[8..15].
// ---------------------------------------------------------------------------
__global__ __launch_bounds__(128) void matvec_wmma_kernel(
    const float* __restrict__ W0, const float* __restrict__ W1,
    const float* __restrict__ W2,
    const float* __restrict__ xbase,
    float* __restrict__ out,            // direct out (gridDim.y==1) or partials
    const float* __restrict__ aux,
    int nrows, int K, int kchunk, int mode, float scale)
{
    __shared__ float sx[4096];
    const int z  = blockIdx.z;
    const int c0 = blockIdx.y * kchunk;
    const float* __restrict__ W = (z == 0) ? W0 : ((z == 1) ? W1 : W2);
    const float* __restrict__ x = xbase + (size_t)z * (size_t)K;

    for (int i = threadIdx.x; i < kchunk; i += 128) sx[i] = x[c0 + i];
    __syncthreads();

    const int wave = threadIdx.x >> 5;
    const int lane = threadIdx.x & 31;
    const int row0 = (blockIdx.x * 4 + wave) * 16;
    if (row0 >= nrows) return;              // wave-uniform: EXEC stays all-1s

    int mrow = row0 + (lane & 15);
    if (mrow > nrows - 1) mrow = nrows - 1; // clamp duplicate loads on tail tile
    const int colsel = (lane >> 4) << 1;    // 0 for lanes 0-15, 2 for lanes 16-31
    const float* __restrict__ wrow = W + (size_t)mrow * (size_t)K + c0 + colsel;
    const float* __restrict__ bp   = sx + colsel;

    v8f acc = {};
#pragma unroll 4
    for (int k = 0; k < kchunk; k += 4) {
        v2f a = *(const v2f*)(wrow + k);    // global_load_b64 (weight stream)
        v2f b = *(const v2f*)(bp + k);      // ds_load_b64 (x broadcast)
        acc = __builtin_amdgcn_wmma_f32_16x16x4_f32(
            /*neg_a=*/false, a, /*neg_b=*/false, b,
            /*c_mod=*/(short)0, acc, /*reuse_a=*/false, /*reuse_b=*/false);
    }

    if ((lane & 15) == 0) {                 // lanes 0 and 16 hold the result
        const int base = row0 + (lane >> 4) * 8;
        if (gridDim.y == 1) {               // direct epilogue (e.g. head matvec)
            float* o = out + (size_t)z * (size_t)nrows;
#pragma unroll
            for (int i = 0; i < 8; i++) {
                int r = base + i;
                if (r < nrows) apply_epilogue(o, aux, r, acc[i], mode, scale);
            }
        } else {                            // partial: part[c][z*nrows + r]
            const size_t nrt = (size_t)gridDim.z * (size_t)nrows;
            float* part = out + (size_t)blockIdx.y * nrt + (size_t)z * (size_t)nrows;
#pragma unroll
            for (int i = 0; i < 8; i++) {
                int r = base + i;
                if (r < nrows) part[r] = acc[i];
            }
        }
    }
}

// Reduce K-chunk partials and apply the epilogue.
__global__ __launch_bounds__(256) void combine_kernel(
    const float* __restrict__ part, float* __restrict__ out,
    const float* __restrict__ aux, int nrt, int nchunks, int mode, float scale)
{
    const int r = blockIdx.x * 256 + threadIdx.x;
    if (r >= nrt) return;
    float y = 0.0f;
    for (int c = 0; c < nchunks; c++) y += part[(size_t)c * nrt + r];
    apply_epilogue(out, aux, r, y, mode, scale);
}

// ---------------------------------------------------------------------------
// LayerNorm (+ optional token-shift mixing) over D=1024, one block of 256.
// ---------------------------------------------------------------------------
__global__ __launch_bounds__(256) void ln_mix_kernel(
    const float* __restrict__ xin, const int* __restrict__ tokenPtr,
    const float* __restrict__ w, const float* __restrict__ b,
    const float* __restrict__ shift,
    const float* __restrict__ m0, const float* __restrict__ m1,
    const float* __restrict__ m2,
    float* __restrict__ ln_out, float* __restrict__ ln_out2,
    float* __restrict__ o0, float* __restrict__ o1, float* __restrict__ o2)
{
    __shared__ float red[256];
    __shared__ float sstat;
    const int t = threadIdx.x;

    const float* xp = xin;
    if (tokenPtr) xp += (size_t)tokenPtr[0] * (size_t)D;

    float v[4];
    float s = 0.0f;
#pragma unroll
    for (int j = 0; j < 4; j++) { v[j] = xp[t + 256 * j]; s += v[j]; }

    red[t] = s;
    __syncthreads();
    for (int off = 128; off > 0; off >>= 1) {
        if (t < off) red[t] += red[t + off];
        __syncthreads();
    }
    if (t == 0) sstat = red[0] * (1.0f / (float)D);
    __syncthreads();
    const float mean = sstat;
    __syncthreads();

    float s2 = 0.0f;
#pragma unroll
    for (int j = 0; j < 4; j++) { float d0 = v[j] - mean; s2 += d0 * d0; }
    red[t] = s2;
    __syncthreads();
    for (int off = 128; off > 0; off >>= 1) {
        if (t < off) red[t] += red[t + off];
        __syncthreads();
    }
    if (t == 0) sstat = red[0] * (1.0f / (float)D);
    __syncthreads();
    const float rstd = rsqrtf(sstat + LN_EPS);

#pragma unroll
    for (int j = 0; j < 4; j++) {
        const int idx = t + 256 * j;
        float y = (v[j] - mean) * rstd * w[idx] + b[idx];
        if (ln_out)  ln_out[idx]  = y;
        if (ln_out2) ln_out2[idx] = y;
        if (o0) { float m = m0[idx]; o0[idx] = y * m + shift[idx] * (1.0f - m); }
        if (o1) { float m = m1[idx]; o1[idx] = y * m + shift[idx] * (1.0f - m); }
        if (o2) { float m = m2[idx]; o2[idx] = y * m + shift[idx] * (1.0f - m); }
    }
}

// ---------------------------------------------------------------------------
// WKV recurrence (elementwise over D): wkv mix + state update + r-gate.
// ---------------------------------------------------------------------------
__global__ __launch_bounds__(256) void wkv_kernel(
    const float* __restrict__ kk, const float* __restrict__ vv,
    const float* __restrict__ rpre,
    const float* __restrict__ tf, const float* __restrict__ td,
    const float* __restrict__ aa_in, const float* __restrict__ bb_in,
    const float* __restrict__ pp_in,
    float* __restrict__ rwkv,
    float* __restrict__ aa_out, float* __restrict__ bb_out,
    float* __restrict__ pp_out)
{
    const int i = blockIdx.x * 256 + threadIdx.x;
    if (i >= D) return;
    const float k  = kk[i], v = vv[i];
    const float aa = aa_in[i], bb = bb_in[i], pp = pp_in[i];

    const float ww = tf[i] + k;
    const float p1 = fmaxf(pp, ww);
    const float e1 = expf(pp - p1), e2 = expf(ww - p1);
    const float wkv = (e1 * aa + e2 * v) / (e1 * bb + e2);

    const float ww2 = pp + td[i];
    const float p2  = fmaxf(ww2, k);
    const float e1b = expf(ww2 - p2), e2b = expf(k - p2);
    aa_out[i] = e1b * aa + e2b * v;
    bb_out[i] = e1b * bb + e2b;
    pp_out[i] = p2;

    const float r = 1.0f / (1.0f + expf(-rpre[i]));
    rwkv[i] = r * wkv;
}

// ---------------------------------------------------------------------------
extern "C" void kernel_launch(void* const* d_in, const int* in_sizes, int n_in,
                              void* d_out, int out_size, void* d_ws, size_t ws_size,
                              hipStream_t stream)
{
    (void)in_sizes; (void)n_in; (void)out_size; (void)ws_size;

    const int*   token = (const int*)d_in[0];
    const float* state = (const float*)d_in[1];
    const float* emb   = (const float*)d_in[2];
    const float* ln0w  = (const float*)d_in[3];
    const float* ln0b  = (const float*)d_in[4];
    const float* ln1w  = (const float*)d_in[5];
    const float* ln1b  = (const float*)d_in[6];
    const float* ln2w  = (const float*)d_in[7];
    const float* ln2b  = (const float*)d_in[8];
    const float* lnow  = (const float*)d_in[9];
    const float* lnob  = (const float*)d_in[10];
    const float* amk   = (const float*)d_in[11];
    const float* amv   = (const float*)d_in[12];
    const float* amr   = (const float*)d_in[13];
    const float* tf    = (const float*)d_in[14];
    const float* td    = (const float*)d_in[15];
    const float* akw   = (const float*)d_in[16];
    const float* avw   = (const float*)d_in[17];
    const float* arw   = (const float*)d_in[18];
    const float* aow   = (const float*)d_in[19];
    const float* fmk   = (const float*)d_in[20];
    const float* fmr   = (const float*)d_in[21];
    const float* fkw   = (const float*)d_in[22];
    const float* fvw   = (const float*)d_in[23];
    const float* frw   = (const float*)d_in[24];
    const float* headw = (const float*)d_in[25];

    float* logits = (float*)d_out;
    float* stout  = (float*)d_out + V;      // new_state follows logits

    float* ws   = (float*)d_ws;
    float* X    = ws;                       // running residual vector
    float* XK   = ws + 1 * D;               // XK/XV/XR contiguous (batched matvec)
    float* XV   = ws + 2 * D;
    float* XR   = ws + 3 * D;
    float* KK   = ws + 4 * D;               // KK/VV/RR contiguous (batched combine)
    float* VVb  = ws + 5 * D;
    float* RR   = ws + 6 * D;
    float* RWKV = ws + 7 * D;
    float* FXK  = ws + 8 * D;
    float* FXR  = ws + 9 * D;
    float* RFP  = ws + 10 * D;
    float* KF   = ws + 11 * D;              // H floats
    float* XLN  = ws + 11 * D + H;
    float* PART = ws + 16 * D;              // K-chunk partials (max 16K floats)

    const dim3 lnb(256), mvb(128);
    const int KCH = 256;                    // K-chunk size (>=256 waves per matvec)

    // split matvec: partial pass over (rows, K-chunks, Z batches) + combine
    auto mvsplit = [&](const float* Wa, const float* Wb, const float* Wc, int Z,
                       const float* x, float* fout, const float* aux,
                       int nrows, int K, int mode, float scale) {
        const int nchunks = K / KCH;
        dim3 g((nrows + 63) / 64, nchunks, Z);
        matvec_wmma_kernel<<<g, mvb, 0, stream>>>(Wa, Wb, Wc, x, PART, nullptr,
                                                  nrows, K, KCH, 0, 1.0f);
        const int nrt = Z * nrows;
        dim3 cg((nrt + 255) / 256);
        combine_kernel<<<cg, lnb, 0, stream>>>(PART, fout, aux, nrt, nchunks,
                                               mode, scale);
    };

    // x = ln0(emb[token])
    ln_mix_kernel<<<1, lnb, 0, stream>>>(emb, token, ln0w, ln0b,
        nullptr, nullptr, nullptr, nullptr, X, nullptr, nullptr, nullptr, nullptr);

    for (int i = 0; i < L; i++) {
        const float* st = state + (size_t)i * 5 * D;
        float*       so = stout + (size_t)i * 5 * D;

        // --- SA: xa = ln1(x); token-shift mixes; new_state[1] = xa
        ln_mix_kernel<<<1, lnb, 0, stream>>>(X, nullptr, ln1w + i * D, ln1b + i * D,
            st + 1 * D, amk + i * D, amv + i * D, amr + i * D,
            nullptr, so + 1 * D, XK, XV, XR);

        // batched k/v/r matvecs (z = 0,1,2): KK = kw@xk, VV = vw@xv, RR = rw@xr
        mvsplit(akw + (size_t)i * D * D, avw + (size_t)i * D * D,
                arw + (size_t)i * D * D, 3, XK, KK, nullptr, D, D, 0, 1.0f);

        wkv_kernel<<<D / 256, lnb, 0, stream>>>(KK, VVb, RR, tf + i * D, td + i * D,
            st + 2 * D, st + 3 * D, st + 4 * D,
            RWKV, so + 2 * D, so + 3 * D, so + 4 * D);

        // x += ow @ (r * wkv)
        mvsplit(aow + (size_t)i * D * D, nullptr, nullptr, 1,
                RWKV, X, nullptr, D, D, 2, 1.0f);

        // --- FFN: xf = ln2(x); mixes; new_state[0] = xf
        ln_mix_kernel<<<1, lnb, 0, stream>>>(X, nullptr, ln2w + i * D, ln2b + i * D,
            st + 0 * D, fmk + i * D, fmr + i * D, nullptr,
            nullptr, so + 0 * D, FXK, FXR, nullptr);

        // kf = relu(fkw @ fxk)^2   (4096 rows)
        mvsplit(fkw + (size_t)i * H * D, nullptr, nullptr, 1,
                FXK, KF, nullptr, H, D, 1, 1.0f);
        // rf_pre = frw @ fxr
        mvsplit(frw + (size_t)i * D * D, nullptr, nullptr, 1,
                FXR, RFP, nullptr, D, D, 0, 1.0f);

        // x = (x + sigmoid(rf_pre) * (fvw @ kf)) * rescale   (K = 4096)
        const float sc = ((i + 1) % 6 == 0) ? 0.5f : 1.0f;
        mvsplit(fvw + (size_t)i * D * H, nullptr, nullptr, 1,
                KF, X, RFP, D, H, 3, sc);
    }

    // logits = head @ lnout(x)  (786 blocks -> direct path, no K-split needed)
    ln_mix_kernel<<<1, lnb, 0, stream>>>(X, nullptr, lnow, lnob,
        nullptr, nullptr, nullptr, nullptr, XLN, nullptr, nullptr, nullptr, nullptr);
    dim3 hg((V + 63) / 64, 1, 1);
    matvec_wmma_kernel<<<hg, mvb, 0, stream>>>(headw, nullptr, nullptr, XLN,
                                               logits, nullptr, V, D, D, 0, 1.0f);
}